// DisjointPNA_76235669504163
// MI455X (gfx1250) — compile-verified
//
#include <hip/hip_runtime.h>
#include <hip/hip_bf16.h>
#include <math.h>

#define N_NODES   50000
#define N_EDGES   1600000
#define CFEAT     32
#define EDGE_DIM  10
#define NUM_TYPES 10
#define NTILES    (N_NODES / 16)   // 3125 exactly
#define WPB       8                // waves per block for GEMM kernels
#define STD_EPS   1e-5f
#define FLT_BIG   3.402823466e38f

typedef __attribute__((ext_vector_type(2))) float v2f;
typedef __attribute__((ext_vector_type(8))) float v8f;

static __device__ __forceinline__ v8f wmma_f32_k4(v2f a, v2f b, v8f c) {
  // D = A(16x4,f32) * B(4x16,f32) + C(16x16,f32)
  return __builtin_amdgcn_wmma_f32_16x16x4_f32(false, a, false, b, (short)0, c, false, false);
}

// ---------------------------------------------------------------------------
// edge type from one-hot edge_attr
__global__ __launch_bounds__(256)
void compute_etype_kernel(const float* __restrict__ eatt, int* __restrict__ etype) {
  int e = blockIdx.x * blockDim.x + threadIdx.x;
  if (e >= N_EDGES) return;
  const float* r = eatt + (size_t)e * EDGE_DIM;
  int t = 0; float best = r[0];
  #pragma unroll
  for (int k = 1; k < EDGE_DIM; ++k) { float v = r[k]; if (v > best) { best = v; t = k; } }
  etype[e] = t;
}

// ---------------------------------------------------------------------------
// reset aggregation buffers
__global__ __launch_bounds__(256)
void init_agg_kernel(float* __restrict__ sum, float* __restrict__ smin,
                     float* __restrict__ smax, float* __restrict__ ssq,
                     float* __restrict__ cnt) {
  int i = blockIdx.x * blockDim.x + threadIdx.x;
  if (i < N_NODES * CFEAT) {
    sum[i] = 0.0f; ssq[i] = 0.0f; smin[i] = FLT_BIG; smax[i] = -FLT_BIG;
  }
  if (i < N_NODES) cnt[i] = 0.0f;
}

// ---------------------------------------------------------------------------
// Y[t][n][:] = x[n][:] @ Wmsg[t] + bmsg[t]   (per-type dense GEMM via f32 WMMA)
// Weights staged once per block into LDS (8 waves share one weight matrix).
__global__ __launch_bounds__(256)
void msg_gemm_kernel(const float* __restrict__ x, const float* __restrict__ Wmsg,
                     const float* __restrict__ bmsg, float* __restrict__ Y) {
  __shared__ float Wl[CFEAT * CFEAT];   // 4 KB
  __shared__ float bl[CFEAT];
  const int t   = blockIdx.y;
  const int tid = threadIdx.x;
  {
    const float* Wg = Wmsg + (size_t)t * CFEAT * CFEAT;
    ((float4*)Wl)[tid] = ((const float4*)Wg)[tid];      // 256 * float4 = 1024 floats
    if (tid < CFEAT) bl[tid] = bmsg[t * CFEAT + tid];
  }
  __syncthreads();

  const int wave  = tid >> 5;
  const int lane  = tid & 31;
  const int ntile = blockIdx.x * WPB + wave;
  if (ntile >= NTILES) return;           // wave-uniform exit (after barrier)
  const int half = lane >> 4;
  const int l    = lane & 15;
  const int m0   = ntile * 16;

  const float b0 = bl[l];
  const float b1 = bl[16 + l];
  v8f acc0, acc1;
  #pragma unroll
  for (int i = 0; i < 8; ++i) { acc0[i] = b0; acc1[i] = b1; }

  const float* xr = x + (size_t)(m0 + l) * CFEAT;
  #pragma unroll
  for (int kc = 0; kc < 8; ++kc) {
    const int kb = kc * 4 + 2 * half;
    v2f a;  a.x  = xr[kb];                    a.y  = xr[kb + 1];
    v2f bb0; bb0.x = Wl[kb * CFEAT + l];      bb0.y = Wl[(kb + 1) * CFEAT + l];
    v2f bb1; bb1.x = Wl[kb * CFEAT + 16 + l]; bb1.y = Wl[(kb + 1) * CFEAT + 16 + l];
    acc0 = wmma_f32_k4(a, bb0, acc0);
    acc1 = wmma_f32_k4(a, bb1, acc1);
  }

  float* Yt = Y + ((size_t)t * N_NODES + m0) * CFEAT;
  #pragma unroll
  for (int i = 0; i < 8; ++i) {
    const int r = i + 8 * half;
    Yt[(size_t)r * CFEAT + l]      = acc0[i];
    Yt[(size_t)r * CFEAT + 16 + l] = acc1[i];
  }
}

// ---------------------------------------------------------------------------
// per-edge scatter into sum/min/max/sumsq/count (8 threads per edge, float4 each)
__global__ __launch_bounds__(256)
void scatter_kernel(const float* __restrict__ Y, const int* __restrict__ etype,
                    const int* __restrict__ eidx,
                    float* __restrict__ sum, float* __restrict__ smin,
                    float* __restrict__ smax, float* __restrict__ ssq,
                    float* __restrict__ cnt) {
  const long long gid = (long long)blockIdx.x * blockDim.x + threadIdx.x;
  const int e = (int)(gid >> 3);
  if (e >= N_EDGES) return;
  const int part = (int)(gid & 7);
  const int s = eidx[e];
  const int d = eidx[N_EDGES + e];
  const int t = etype[e];
  const float4 m = *((const float4*)(Y + ((size_t)t * N_NODES + s) * CFEAT) + part);
  const size_t base = (size_t)d * CFEAT + (size_t)part * 4;

  atomicAdd(sum + base + 0, m.x);
  atomicAdd(sum + base + 1, m.y);
  atomicAdd(sum + base + 2, m.z);
  atomicAdd(sum + base + 3, m.w);
  atomicAdd(ssq + base + 0, m.x * m.x);
  atomicAdd(ssq + base + 1, m.y * m.y);
  atomicAdd(ssq + base + 2, m.z * m.z);
  atomicAdd(ssq + base + 3, m.w * m.w);
  __hip_atomic_fetch_min(smin + base + 0, m.x, __ATOMIC_RELAXED, __HIP_MEMORY_SCOPE_AGENT);
  __hip_atomic_fetch_min(smin + base + 1, m.y, __ATOMIC_RELAXED, __HIP_MEMORY_SCOPE_AGENT);
  __hip_atomic_fetch_min(smin + base + 2, m.z, __ATOMIC_RELAXED, __HIP_MEMORY_SCOPE_AGENT);
  __hip_atomic_fetch_min(smin + base + 3, m.w, __ATOMIC_RELAXED, __HIP_MEMORY_SCOPE_AGENT);
  __hip_atomic_fetch_max(smax + base + 0, m.x, __ATOMIC_RELAXED, __HIP_MEMORY_SCOPE_AGENT);
  __hip_atomic_fetch_max(smax + base + 1, m.y, __ATOMIC_RELAXED, __HIP_MEMORY_SCOPE_AGENT);
  __hip_atomic_fetch_max(smax + base + 2, m.z, __ATOMIC_RELAXED, __HIP_MEMORY_SCOPE_AGENT);
  __hip_atomic_fetch_max(smax + base + 3, m.w, __ATOMIC_RELAXED, __HIP_MEMORY_SCOPE_AGENT);
  if (part == 0) atomicAdd(cnt + d, 1.0f);
}

// ---------------------------------------------------------------------------
// agg[n] = concat(sum, min, max, std)  with empty-segment masking
__global__ __launch_bounds__(256)
void finalize_kernel(const float* __restrict__ sum, const float* __restrict__ smin,
                     const float* __restrict__ smax, const float* __restrict__ ssq,
                     const float* __restrict__ cnt, float* __restrict__ agg) {
  int i = blockIdx.x * blockDim.x + threadIdx.x;
  if (i >= N_NODES * CFEAT) return;
  const int n = i >> 5;
  const int c = i & 31;
  const float ct   = cnt[n];
  const float s    = sum[i];
  const float cc   = fmaxf(ct, 1.0f);
  const float mean = s / cc;
  const float var  = fmaxf(ssq[i] / cc - mean * mean, 0.0f);
  float stdv = sqrtf(var + STD_EPS);
  const bool has = ct > 0.0f;
  const float mn = has ? smin[i] : 0.0f;
  const float mx = has ? smax[i] : 0.0f;
  stdv = has ? stdv : 0.0f;
  float* a = agg + (size_t)n * (4 * CFEAT);
  a[c]             = s;
  a[CFEAT + c]     = mn;
  a[2 * CFEAT + c] = mx;
  a[3 * CFEAT + c] = stdv;
}

// ---------------------------------------------------------------------------
// xout[n] = act(agg[n] @ Wnode[t] + bnode[t]) + (x[n] @ Wres[t] + bres[t])
// for nodes with node_ids[n] == t (grid-Y over types, selective writeback).
// Both weight matrices staged into LDS once per block (shared by 8 waves).
__global__ __launch_bounds__(256)
void node_update_kernel(const float* __restrict__ agg, const float* __restrict__ xin,
                        const float* __restrict__ Wnode, const float* __restrict__ bnode,
                        const float* __restrict__ Wres, const float* __restrict__ bres,
                        const int* __restrict__ node_ids, float* __restrict__ xout,
                        const int apply_relu) {
  __shared__ float Wnl[4 * CFEAT * CFEAT];  // 16 KB
  __shared__ float Wrl[CFEAT * CFEAT];      //  4 KB
  __shared__ float bnl[CFEAT], brl[CFEAT];
  const int t   = blockIdx.y;
  const int tid = threadIdx.x;
  {
    const float4* Wng = (const float4*)(Wnode + (size_t)t * (4 * CFEAT) * CFEAT);
    #pragma unroll
    for (int i = 0; i < 4; ++i) ((float4*)Wnl)[tid + 256 * i] = Wng[tid + 256 * i];
    ((float4*)Wrl)[tid] = ((const float4*)(Wres + (size_t)t * CFEAT * CFEAT))[tid];
    if (tid < CFEAT) { bnl[tid] = bnode[t * CFEAT + tid]; brl[tid] = bres[t * CFEAT + tid]; }
  }
  __syncthreads();

  const int wave  = tid >> 5;
  const int lane  = tid & 31;
  const int ntile = blockIdx.x * WPB + wave;
  if (ntile >= NTILES) return;           // wave-uniform (after barrier)
  const int half = lane >> 4;
  const int l    = lane & 15;
  const int m0   = ntile * 16;

  // skip tiles with no node of this type (ballot is wave-uniform -> EXEC stays full)
  unsigned long long mask = __ballot(node_ids[m0 + l] == t);
  if ((mask & 0xFFFFFFFFull) == 0ull) return;

  const float bn0 = bnl[l], bn1 = bnl[16 + l];
  const float br0 = brl[l], br1 = brl[16 + l];

  v8f a0, a1, r0, r1;
  #pragma unroll
  for (int i = 0; i < 8; ++i) { a0[i] = bn0; a1[i] = bn1; r0[i] = br0; r1[i] = br1; }

  // conv part: [16 x 128] @ [128 x 32]
  const float* ar = agg + (size_t)(m0 + l) * (4 * CFEAT);
  #pragma unroll
  for (int kc = 0; kc < 32; ++kc) {
    const int kb = kc * 4 + 2 * half;
    v2f a;  a.x  = ar[kb];                    a.y  = ar[kb + 1];
    v2f b0; b0.x = Wnl[kb * CFEAT + l];       b0.y = Wnl[(kb + 1) * CFEAT + l];
    v2f b1; b1.x = Wnl[kb * CFEAT + 16 + l];  b1.y = Wnl[(kb + 1) * CFEAT + 16 + l];
    a0 = wmma_f32_k4(a, b0, a0);
    a1 = wmma_f32_k4(a, b1, a1);
  }
  // residual part: [16 x 32] @ [32 x 32]
  const float* xr = xin + (size_t)(m0 + l) * CFEAT;
  #pragma unroll
  for (int kc = 0; kc < 8; ++kc) {
    const int kb = kc * 4 + 2 * half;
    v2f a;  a.x  = xr[kb];                    a.y  = xr[kb + 1];
    v2f b0; b0.x = Wrl[kb * CFEAT + l];       b0.y = Wrl[(kb + 1) * CFEAT + l];
    v2f b1; b1.x = Wrl[kb * CFEAT + 16 + l];  b1.y = Wrl[(kb + 1) * CFEAT + 16 + l];
    r0 = wmma_f32_k4(a, b0, r0);
    r1 = wmma_f32_k4(a, b1, r1);
  }

  float* xo = xout + (size_t)m0 * CFEAT;
  #pragma unroll
  for (int i = 0; i < 8; ++i) {
    const int r = i + 8 * half;
    if (node_ids[m0 + r] == t) {
      float h0 = a0[i], h1 = a1[i];
      if (apply_relu) { h0 = fmaxf(h0, 0.0f); h1 = fmaxf(h1, 0.0f); }
      xo[(size_t)r * CFEAT + l]      = h0 + r0[i];
      xo[(size_t)r * CFEAT + 16 + l] = h1 + r1[i];
    }
  }
}

// ---------------------------------------------------------------------------
extern "C" void kernel_launch(void* const* d_in, const int* in_sizes, int n_in,
                              void* d_out, int out_size, void* d_ws, size_t ws_size,
                              hipStream_t stream) {
  const float* x0   = (const float*)d_in[0];
  const float* eatt = (const float*)d_in[1];
  const int*   eidx = (const int*)d_in[2];
  const int*   nids = (const int*)d_in[3];
  const float* Wmsg[2]  = {(const float*)d_in[4],  (const float*)d_in[10]};
  const float* bmsg[2]  = {(const float*)d_in[5],  (const float*)d_in[11]};
  const float* Wnode[2] = {(const float*)d_in[6],  (const float*)d_in[12]};
  const float* bnode[2] = {(const float*)d_in[7],  (const float*)d_in[13]};
  const float* Wres[2]  = {(const float*)d_in[8],  (const float*)d_in[14]};
  const float* bres[2]  = {(const float*)d_in[9],  (const float*)d_in[15]};

  char* p = (char*)d_ws;
  float* Y    = (float*)p; p += sizeof(float) * (size_t)EDGE_DIM * N_NODES * CFEAT; // 64 MB
  float* sum  = (float*)p; p += sizeof(float) * (size_t)N_NODES * CFEAT;
  float* smin = (float*)p; p += sizeof(float) * (size_t)N_NODES * CFEAT;
  float* smax = (float*)p; p += sizeof(float) * (size_t)N_NODES * CFEAT;
  float* ssq  = (float*)p; p += sizeof(float) * (size_t)N_NODES * CFEAT;
  float* cnt  = (float*)p; p += sizeof(float) * (size_t)N_NODES;
  float* agg  = (float*)p; p += sizeof(float) * (size_t)N_NODES * 4 * CFEAT;
  float* x1   = (float*)p; p += sizeof(float) * (size_t)N_NODES * CFEAT;
  int* etype  = (int*)p;   p += sizeof(int)   * (size_t)N_EDGES;

  compute_etype_kernel<<<(N_EDGES + 255) / 256, 256, 0, stream>>>(eatt, etype);

  const dim3 ggrid((NTILES + WPB - 1) / WPB, EDGE_DIM);
  const dim3 ngrid((NTILES + WPB - 1) / WPB, NUM_TYPES);
  const long long scat_threads = (long long)N_EDGES * 8;

  const float* xin = x0;
  for (int step = 0; step < 2; ++step) {
    float* xout = (step == 0) ? x1 : (float*)d_out;

    init_agg_kernel<<<(N_NODES * CFEAT + 255) / 256, 256, 0, stream>>>(sum, smin, smax, ssq, cnt);
    msg_gemm_kernel<<<ggrid, 256, 0, stream>>>(xin, Wmsg[step], bmsg[step], Y);
    scatter_kernel<<<(unsigned)((scat_threads + 255) / 256), 256, 0, stream>>>(
        Y, etype, eidx, sum, smin, smax, ssq, cnt);
    finalize_kernel<<<(N_NODES * CFEAT + 255) / 256, 256, 0, stream>>>(sum, smin, smax, ssq, cnt, agg);
    node_update_kernel<<<ngrid, 256, 0, stream>>>(
        agg, xin, Wnode[step], bnode[step], Wres[step], bres[step], nids, xout,
        step == 0 ? 1 : 0);
    xin = xout;
  }
}